// SpatialTransformer_11879879541358
// MI455X (gfx1250) — compile-verified
//
#include <hip/hip_runtime.h>
#include <cstdint>

// Problem dims (fixed by the reference setup)
#define B_DIM   4
#define C_DIM   64
#define H_DIM   256
#define W_DIM   512
#define TPB     256   // 8 wave32 per block
#define STAGE_C 8     // channels per TDM tile (16 KB)
#define NBUF    3     // triple-buffered stage -> single barrier per stage
#define NSTAGE  (C_DIM / STAGE_C)

typedef uint32_t u32x4 __attribute__((ext_vector_type(4)));
typedef uint32_t u32x8 __attribute__((ext_vector_type(8)));

// One block per (b,h). Warp weights live in registers (each thread owns the
// same two width positions for all 64 channels). Channel rows stream through
// a triple-buffered LDS stage, 8 channels (16 KB) per Tensor-Data-Mover
// descriptor (tensor_load_to_lds / TENSORcnt), with the next tiles' DMA
// overlapping the current tile's gather+lerp. Both lerp taps come from one
// ds_load_2addr_b32 (taps are adjacent; the clamped-vs-adjacent difference is
// always weight-masked to zero).
__global__ __launch_bounds__(TPB) void disparity_warp_kernel(
    const float* __restrict__ right,   // [B,C,H,W]
    const float* __restrict__ disp,    // [B,1,H,W]
    float* __restrict__ out)           // [B,C,H,W]
{
    __shared__ float s_stage[NBUF][STAGE_C][W_DIM];   // 3 x 16 KB staging ring

    const int tid = threadIdx.x;
    const int bh  = blockIdx.x;          // 0 .. B*H-1
    const int h   = bh % H_DIM;
    const int b   = bh / H_DIM;

    const size_t   chan_stride = (size_t)H_DIM * W_DIM;  // elems between channels
    const uint64_t base_addr   =
        (uint64_t)(uintptr_t)(right + ((size_t)b * C_DIM * H_DIM + h) * W_DIM);
    float* obase = out + ((size_t)b * C_DIM * H_DIM + h) * W_DIM;

    const bool issuer = (tid < 32);   // wave 0 drives the TDM

    // One TDM descriptor: 2-D tile, STAGE_C rows x 512 f32, row stride H*W.
    auto tdm_issue = [&](int stage, int buf) {
        const uint64_t ga = base_addr +
            (uint64_t)(stage * STAGE_C) * chan_stride * sizeof(float);
        const uint32_t lds_base = (uint32_t)(uintptr_t)(&s_stage[buf][0][0]);
        u32x4 g0;
        g0[0] = 1u;                                   // count=1 (valid user D#)
        g0[1] = lds_base;                             // lds_addr
        g0[2] = (uint32_t)ga;                         // global_addr lo
        g0[3] = (uint32_t)(ga >> 32) | (2u << 30);    // global_addr hi | type=2
        u32x8 g1;
        g1[0] = 2u << 16;                             // data_size = 4 B
        g1[1] = (uint32_t)W_DIM << 16;                // tensor_dim0 = 512 (lo16)
        g1[2] = 0;                                    // tensor_dim0 hi | tensor_dim1 lo
        g1[3] = ((uint32_t)W_DIM << 16) | 0x10u;      // tile_dim0=512 | tensor_dim1 hi (=1M)
        g1[4] = (uint32_t)STAGE_C;                    // tile_dim1 = 8, tile_dim2 = 0
        g1[5] = (uint32_t)(H_DIM * W_DIM);            // tensor_dim0_stride lo32 = 131072
        g1[6] = 0;                                    // stride hi | dim1_stride lo
        g1[7] = 0;
        asm volatile("tensor_load_to_lds %0, %1" :: "s"(g0), "s"(g1) : "memory");
    };

    // Kick off the first two tiles before doing any ALU work.
    if (issuer) {
        tdm_issue(0, 0);
        tdm_issue(1, 1);
    }

    // ---- Per-thread warp weights (registers; overlap with stage-0 DMA) ----
    const float* disp_row = disp + ((size_t)b * H_DIM + h) * W_DIM;
    float r_wa[2], r_wb[2];
    int   r_ia[2];
    #pragma unroll
    for (int k = 0; k < W_DIM / TPB; ++k) {
        const int w = tid + k * TPB;
        const float d  = disp_row[w];
        const float ry = (float)w + d;
        const float ra = floorf(ry);
        const float wb = ry - ra;
        const float wa = 1.0f - wb;
        int ia = (int)ra;
        ia = min(max(ia, 0), W_DIM - 1);
        const float mask = (ry < 0.0f || ry > (float)(W_DIM - 1)) ? 0.0f : 1.0f;
        r_wa[k] = wa * mask;
        r_wb[k] = wb * mask;
        r_ia[k] = ia;
    }

    for (int s = 0; s < NSTAGE; ++s) {
        // Outstanding TDM ops: stages s and s+1 (s+2 not yet issued), so
        // <=1 pins stage s; on the last stage drain fully.
        if (s + 1 < NSTAGE) __builtin_amdgcn_s_wait_tensorcnt(1);
        else                __builtin_amdgcn_s_wait_tensorcnt(0);
        __syncthreads();   // tile s visible to all waves; also proves all waves
                           // finished reading buffer (s+2)%NBUF during stage s-1

        if (issuer && (s + 2 < NSTAGE)) tdm_issue(s + 2, (s + 2) % NBUF);

        const int cur = s % NBUF;
        #pragma unroll
        for (int cc = 0; cc < STAGE_C; ++cc) {
            const float* row  = s_stage[cur][cc];
            float*       orow = obase + (size_t)(s * STAGE_C + cc) * chan_stride;
            #pragma unroll
            for (int k = 0; k < W_DIM / TPB; ++k) {
                const int w  = tid + k * TPB;
                const int ia = r_ia[k];
                const float fa = row[ia];        // merged by backend into one
                const float fb = row[ia + 1];    // ds_load_2addr_b32 (off 0,1)
                const float v  = r_wa[k] * fa + r_wb[k] * fb;
                __builtin_nontemporal_store(v, &orow[w]);  // write-once output
            }
        }
    }
}

extern "C" void kernel_launch(void* const* d_in, const int* in_sizes, int n_in,
                              void* d_out, int out_size, void* d_ws, size_t ws_size,
                              hipStream_t stream) {
    (void)in_sizes; (void)n_in; (void)out_size; (void)d_ws; (void)ws_size;
    const float* right = (const float*)d_in[0];
    const float* disp  = (const float*)d_in[1];
    float*       out   = (float*)d_out;

    dim3 grid(B_DIM * H_DIM);   // 1024 blocks, one per (b,h)
    dim3 block(TPB);
    disparity_warp_kernel<<<grid, block, 0, stream>>>(right, disp, out);
}